// PD_HAN_2413771621182
// MI455X (gfx1250) — compile-verified
//
#include <hip/hip_runtime.h>
#include <hip/hip_bf16.h>

typedef __attribute__((ext_vector_type(2))) float v2f;
typedef __attribute__((ext_vector_type(8))) float v8f;

constexpr int B   = 16;
constexpr int N   = 512;
constexpr int D   = 256;
constexpr int PH  = 8;
constexpr int DHP = 32;     // D / PH
constexpr int NP1 = 513;    // N + 1
constexpr int UH  = 4;
constexpr int DHU = 64;     // D / UH
constexpr int FH  = 128;
constexpr int NC  = 2;

// ---------------------------------------------------------------------------
// Y = X(MxK) @ W^T(Nout x K) + bias, f32 WMMA 16x16x4.
// One wave per 16x32 output tile (two col tiles share the A fragment).
// REQUIRES: M % 16 == 0, Nout % 32 == 0, K % 8 == 0 (true for every call).
// K is consumed in permuted chunks of 8: lane-half 0 owns k0..k0+3 (one b128
// load), lane-half 1 owns k0+4..k0+7; elements {0,1} feed one WMMA and {2,3}
// the next. The same permutation is applied to A and B, so the dot product
// is unchanged. ldy = output row stride (for scatter into u).
// ---------------------------------------------------------------------------
__global__ void gemm16_wmma(const float* __restrict__ X,
                            const float* __restrict__ W,
                            const float* __restrict__ bias,
                            float* __restrict__ Y,
                            int M, int K, int ldy) {
    const int lane = threadIdx.x;
    const int half = lane >> 4;
    const int mrow = lane & 15;
    const int nl   = lane & 15;
    const int m    = blockIdx.x * 16 + mrow;
    const int col0 = blockIdx.y * 32 + nl;
    const int col1 = col0 + 16;

    v8f acc0, acc1;
    const float bi0 = bias ? bias[col0] : 0.0f;
    const float bi1 = bias ? bias[col1] : 0.0f;
#pragma unroll
    for (int i = 0; i < 8; ++i) { acc0[i] = bi0; acc1[i] = bi1; }

    const float* xrow = X + (size_t)m * K;
    const float* w0   = W + (size_t)col0 * K;
    const float* w1   = W + (size_t)col1 * K;

    for (int k0 = 0; k0 < K; k0 += 8) {
        const int ka = k0 + half * 4;
        const float4 ax = *(const float4*)(xrow + ka);
        const float4 bx = *(const float4*)(w0 + ka);
        const float4 cx = *(const float4*)(w1 + ka);

        v2f aLo; aLo[0] = ax.x; aLo[1] = ax.y;
        v2f aHi; aHi[0] = ax.z; aHi[1] = ax.w;
        v2f bLo; bLo[0] = bx.x; bLo[1] = bx.y;
        v2f bHi; bHi[0] = bx.z; bHi[1] = bx.w;
        v2f cLo; cLo[0] = cx.x; cLo[1] = cx.y;
        v2f cHi; cHi[0] = cx.z; cHi[1] = cx.w;

        acc0 = __builtin_amdgcn_wmma_f32_16x16x4_f32(
            false, aLo, false, bLo, (short)0, acc0, false, false);
        acc1 = __builtin_amdgcn_wmma_f32_16x16x4_f32(
            false, aLo, false, cLo, (short)0, acc1, false, false);
        acc0 = __builtin_amdgcn_wmma_f32_16x16x4_f32(
            false, aHi, false, bHi, (short)0, acc0, false, false);
        acc1 = __builtin_amdgcn_wmma_f32_16x16x4_f32(
            false, aHi, false, cHi, (short)0, acc1, false, false);
    }

#pragma unroll
    for (int i = 0; i < 8; ++i) {
        const int row = blockIdx.x * 16 + i + 8 * half;
        Y[(size_t)row * ldy + col0] = acc0[i];
        Y[(size_t)row * ldy + col1] = acc1[i];
    }
}

// ---------------------------------------------------------------------------
// P_GAT per-head attention logit halves: sq/sk from qk and attn weights.
// Layout sq/sk: [(b*N + n)*PH + h]
// ---------------------------------------------------------------------------
__global__ void pgat_sqsk(const float* __restrict__ qk,
                          const float* __restrict__ attn,
                          float* __restrict__ sq,
                          float* __restrict__ sk) {
    const int idx = blockIdx.x * blockDim.x + threadIdx.x;
    if (idx >= B * N * PH) return;
    const int h = idx % PH;
    const int n = (idx / PH) % N;
    const int b = idx / (PH * N);
    const float* v = qk + ((size_t)b * N + n) * D + h * DHP;
    const float* wq = attn + h * (2 * DHP);
    const float* wk = wq + DHP;
    float aq = 0.f, ak = 0.f;
#pragma unroll 8
    for (int d = 0; d < DHP; ++d) {
        const float x = v[d];
        aq += x * wq[d];
        ak += x * wk[d];
    }
    sq[idx] = aq;
    sk[idx] = ak;
}

// ---------------------------------------------------------------------------
// Fused P_GAT attention: softmax(tanh(sq+sk)) @ qk_head, +residual, tanh.
// One wave per (b, h, 16-row q tile). A-fragment = unnormalized exp weights
// built in-register (tanh-bounded scores: no max-subtraction needed);
// Z accumulated in VALU, lane halves merged via shfl_xor(16) (wave32).
// ---------------------------------------------------------------------------
__global__ void pgat_attn_step(const float* __restrict__ cur,
                               const float* __restrict__ qk,
                               const float* __restrict__ sq,
                               const float* __restrict__ sk,
                               const float* __restrict__ pm,
                               float* __restrict__ out) {
    const int b  = blockIdx.z;
    const int h  = blockIdx.y;
    const int qt = blockIdx.x * 16;
    const int lane = threadIdx.x;
    const int half = lane >> 4;
    const int mrow = lane & 15;
    const int nl   = lane & 15;

    const int q = qt + mrow;
    const float sql = sq[((size_t)b * N + q) * PH + h];
    const float mq  = pm[(size_t)b * N + q];

    v8f acc0, acc1;
#pragma unroll
    for (int i = 0; i < 8; ++i) { acc0[i] = 0.f; acc1[i] = 0.f; }
    float zp = 0.f;

    for (int k0 = 0; k0 < N; k0 += 4) {
        const int k = k0 + half * 2;
        const float mk0 = pm[(size_t)b * N + k];
        const float mk1 = pm[(size_t)b * N + k + 1];
        const float s0 = (mq * mk0 == 0.f) ? -1e-8f
                         : tanhf(sql + sk[((size_t)b * N + k) * PH + h]);
        const float s1 = (mq * mk1 == 0.f) ? -1e-8f
                         : tanhf(sql + sk[((size_t)b * N + k + 1) * PH + h]);
        const float w0 = __expf(s0);
        const float w1 = __expf(s1);
        zp += w0 + w1;
        v2f a; a[0] = w0; a[1] = w1;

        const float* r0 = qk + ((size_t)b * N + k) * D + h * DHP;
        const float* r1 = r0 + D;
        v2f b0; b0[0] = r0[nl];      b0[1] = r1[nl];
        v2f b1; b1[0] = r0[16 + nl]; b1[1] = r1[16 + nl];
        acc0 = __builtin_amdgcn_wmma_f32_16x16x4_f32(
            false, a, false, b0, (short)0, acc0, false, false);
        acc1 = __builtin_amdgcn_wmma_f32_16x16x4_f32(
            false, a, false, b1, (short)0, acc1, false, false);
    }

    // Row normalizers: lane l and l^16 cover complementary K subsets of row mrow.
    const float z = zp + __shfl_xor(zp, 16, 32);
    __shared__ float zs[16];
    if (lane < 16) zs[mrow] = z;
    __syncthreads();

#pragma unroll
    for (int i = 0; i < 8; ++i) {
        const int row = qt + i + 8 * half;
        const float invz = 1.0f / zs[i + 8 * half];
        const size_t base = ((size_t)b * N + row) * D + h * DHP;
        out[base + nl]      = cur[base + nl]      + tanhf(acc0[i] * invz);
        out[base + 16 + nl] = cur[base + 16 + nl] + tanhf(acc1[i] * invz);
    }
}

// ---------------------------------------------------------------------------
// comb = concat(new_p, c_node) along node axis -> (B, 513, D)
// ---------------------------------------------------------------------------
__global__ void build_comb(const float* __restrict__ feat,
                           const float* __restrict__ c_node,
                           float* __restrict__ comb) {
    const int idx = blockIdx.x * blockDim.x + threadIdx.x;
    if (idx >= B * NP1 * D) return;
    const int d = idx % D;
    const int r = (idx / D) % NP1;
    const int b = idx / (D * NP1);
    comb[idx] = (r < N) ? feat[((size_t)b * N + r) * D + d]
                        : c_node[(size_t)b * D + d];
}

// sim[b,k] = dot(q[b], kbuf[b,k]) / 16, masked
__global__ void up_sim(const float* __restrict__ qbuf,
                       const float* __restrict__ kbuf,
                       const float* __restrict__ pm,
                       float* __restrict__ sim) {
    const int idx = blockIdx.x * blockDim.x + threadIdx.x;
    if (idx >= B * NP1) return;
    const int k = idx % NP1;
    const int b = idx / NP1;
    const float* qv = qbuf + (size_t)b * D;
    const float* kv = kbuf + (size_t)idx * D;
    float acc = 0.f;
#pragma unroll 8
    for (int d = 0; d < D; ++d) acc += qv[d] * kv[d];
    const float m = (k < N) ? pm[(size_t)b * N + k] : 1.0f;
    sim[idx] = (m == 0.f) ? 1e-8f : acc * (1.0f / 16.0f);
}

// In-place softmax over 513 keys per batch row (max-subtracted).
__global__ void up_softmax(float* __restrict__ sim) {
    __shared__ float red[256];
    const int b = blockIdx.x;
    const int t = threadIdx.x;
    float* row = sim + (size_t)b * NP1;

    float m = -3.4e38f;
    for (int k = t; k < NP1; k += 256) m = fmaxf(m, row[k]);
    red[t] = m; __syncthreads();
    for (int s = 128; s > 0; s >>= 1) {
        if (t < s) red[t] = fmaxf(red[t], red[t + s]);
        __syncthreads();
    }
    const float mx = red[0]; __syncthreads();

    float acc = 0.f;
    for (int k = t; k < NP1; k += 256) {
        const float e = __expf(row[k] - mx);
        row[k] = e;
        acc += e;
    }
    red[t] = acc; __syncthreads();
    for (int s = 128; s > 0; s >>= 1) {
        if (t < s) red[t] += red[t + s];
        __syncthreads();
    }
    const float inv = 1.0f / red[0]; __syncthreads();
    for (int k = t; k < NP1; k += 256) row[k] *= inv;
}

// head[b,d] = sum_k att[b,k] * comb[b,k,d]
__global__ void up_head(const float* __restrict__ att,
                        const float* __restrict__ comb,
                        float* __restrict__ head) {
    const int idx = blockIdx.x * blockDim.x + threadIdx.x;
    if (idx >= B * D) return;
    const int d = idx % D;
    const int b = idx / D;
    const float* av = att + (size_t)b * NP1;
    const float* cv = comb + (size_t)b * NP1 * D + d;
    float acc = 0.f;
    for (int k = 0; k < NP1; ++k) acc += av[k] * cv[(size_t)k * D];
    head[idx] = acc;
}

// ow_eff[d][d'] = sum_{r<4} ow[d][r*D + d']   (identical-head collapse)
__global__ void ow_eff_k(const float* __restrict__ ow, float* __restrict__ eff) {
    const int idx = blockIdx.x * blockDim.x + threadIdx.x;
    if (idx >= D * D) return;
    const int dp = idx % D;
    const int d  = idx / D;
    const float* r = ow + (size_t)d * (D * 4) + dp;
    eff[idx] = r[0] + r[D] + r[2 * D] + r[3 * D];
}

// U_GAT sq/sk (4 heads, dh=64), rows = (b*4 + n)
__global__ void ugat_sqsk(const float* __restrict__ qku,
                          const float* __restrict__ attn,
                          float* __restrict__ sq,
                          float* __restrict__ sk) {
    const int idx = blockIdx.x * blockDim.x + threadIdx.x;
    if (idx >= B * 4 * UH) return;
    const int h = idx % UH;
    const int n = (idx / UH) % 4;
    const int b = idx / (UH * 4);
    const float* v  = qku + ((size_t)b * 4 + n) * D + h * DHU;
    const float* wq = attn + h * (2 * DHU);
    const float* wk = wq + DHU;
    float aq = 0.f, ak = 0.f;
#pragma unroll 8
    for (int d = 0; d < DHU; ++d) {
        const float x = v[d];
        aq += x * wq[d];
        ak += x * wk[d];
    }
    sq[idx] = aq;
    sk[idx] = ak;
}

// U_GAT attention over K=4 users, + residual tanh. Thread per (b,q,h,d).
__global__ void u_attn(const float* __restrict__ u,
                       const float* __restrict__ qku,
                       const float* __restrict__ sq,
                       const float* __restrict__ sk,
                       float* __restrict__ u2) {
    const int idx = blockIdx.x * blockDim.x + threadIdx.x;
    if (idx >= B * 4 * UH * DHU) return;
    const int d = idx % DHU;
    const int h = (idx / DHU) % UH;
    const int q = (idx / (DHU * UH)) % 4;
    const int b = idx / (DHU * UH * 4);
    const float sql = sq[((size_t)b * 4 + q) * UH + h];
    float z = 0.f, acc = 0.f;
#pragma unroll
    for (int k = 0; k < 4; ++k) {
        const float s = tanhf(sql + sk[((size_t)b * 4 + k) * UH + h]);
        const float w = __expf(s);
        z += w;
        acc += w * qku[((size_t)b * 4 + k) * D + h * DHU + d];
    }
    const size_t o = ((size_t)b * 4 + q) * D + h * DHU + d;
    u2[o] = u[o] + tanhf(acc / z);
}

// h = relu(x @ W^T + b); rows of x are u2[(b*4+i)*D]; out rows (i*B+b)*FH
__global__ void mlp1_k(const float* __restrict__ u2,
                       const float* __restrict__ w, const float* __restrict__ bias,
                       float* __restrict__ h1) {
    const int idx = blockIdx.x * blockDim.x + threadIdx.x;
    if (idx >= 4 * B * FH) return;
    const int f = idx % FH;
    const int b = (idx / FH) % B;
    const int i = idx / (FH * B);
    const float* x  = u2 + ((size_t)b * 4 + i) * D;
    const float* wr = w + (size_t)f * D;
    float acc = bias[f];
#pragma unroll 8
    for (int d = 0; d < D; ++d) acc += x[d] * wr[d];
    h1[((size_t)i * B + b) * FH + f] = fmaxf(acc, 0.f);
}

__global__ void mlp2_k(const float* __restrict__ h1,
                       const float* __restrict__ w, const float* __restrict__ bias,
                       float* __restrict__ h2) {
    const int idx = blockIdx.x * blockDim.x + threadIdx.x;
    if (idx >= 4 * B * FH) return;
    const int f = idx % FH;
    const int r = idx / FH;                       // r = i*B + b
    const float* x  = h1 + (size_t)r * FH;
    const float* wr = w + (size_t)f * FH;
    float acc = bias[f];
#pragma unroll 8
    for (int g = 0; g < FH; ++g) acc += x[g] * wr[g];
    h2[(size_t)r * FH + f] = fmaxf(acc, 0.f);
}

__global__ void heads_k(const float* __restrict__ h2,
                        const float* __restrict__ w0, const float* __restrict__ b0,
                        const float* __restrict__ w1, const float* __restrict__ b1,
                        const float* __restrict__ w2, const float* __restrict__ b2,
                        const float* __restrict__ w3, const float* __restrict__ b3,
                        float* __restrict__ out) {
    const int idx = threadIdx.x;
    if (idx >= 4 * B * NC) return;
    const int c = idx % NC;
    const int b = (idx / NC) % B;
    const int i = idx / (NC * B);
    const float* ws[4] = {w0, w1, w2, w3};
    const float* bs[4] = {b0, b1, b2, b3};
    const float* x  = h2 + ((size_t)i * B + b) * FH;
    const float* wr = ws[i] + (size_t)c * FH;
    float acc = bs[i][c];
#pragma unroll 8
    for (int g = 0; g < FH; ++g) acc += x[g] * wr[g];
    out[idx] = acc;
}

// ---------------------------------------------------------------------------
extern "C" void kernel_launch(void* const* d_in, const int* in_sizes, int n_in,
                              void* d_out, int out_size, void* d_ws, size_t ws_size,
                              hipStream_t stream) {
    const float* p_mask    = (const float*)d_in[0];
    const float* feature   = (const float*)d_in[1];
    const float* c_node    = (const float*)d_in[2];
    const float* pgat_W    = (const float*)d_in[3];
    const float* pgat_attn = (const float*)d_in[4];
    const float *up_qw[4], *up_qb[4], *up_kw[4], *up_kb[4], *up_ow[4], *up_ob[4];
    for (int i = 0; i < 4; ++i) {
        const int base = 5 + i * 6;
        up_qw[i] = (const float*)d_in[base + 0];
        up_qb[i] = (const float*)d_in[base + 1];
        up_kw[i] = (const float*)d_in[base + 2];
        up_kb[i] = (const float*)d_in[base + 3];
        up_ow[i] = (const float*)d_in[base + 4];
        up_ob[i] = (const float*)d_in[base + 5];
    }
    const float* ugat_W    = (const float*)d_in[29];
    const float* ugat_attn = (const float*)d_in[30];
    const float* mlp_w1    = (const float*)d_in[31];
    const float* mlp_b1    = (const float*)d_in[32];
    const float* mlp_w2    = (const float*)d_in[33];
    const float* mlp_b2    = (const float*)d_in[34];
    const float *fcf_w[4], *fcf_b[4];
    for (int i = 0; i < 4; ++i) {
        fcf_w[i] = (const float*)d_in[35 + 2 * i];
        fcf_b[i] = (const float*)d_in[36 + 2 * i];
    }
    float* out = (float*)d_out;

    // Workspace layout (floats); all offsets are multiples of 16 floats so
    // every row base stays 16B-aligned for the b128 load path.
    float* ws = (float*)d_ws;
    size_t off = 0;
    float* feat0 = ws + off; off += (size_t)B * N * D;
    float* feat1 = ws + off; off += (size_t)B * N * D;
    float* qk    = ws + off; off += (size_t)B * NP1 * D;   // doubles as kbuf
    float* sq    = ws + off; off += (size_t)B * N * PH;
    float* sk    = ws + off; off += (size_t)B * N * PH;
    float* comb  = ws + off; off += (size_t)B * NP1 * D;
    float* qbuf  = ws + off; off += (size_t)B * D;
    float* oweff = ws + off; off += (size_t)D * D;
    float* sim   = ws + off; off += (size_t)B * NP1;       // 8208 = 513*16
    float* headb = ws + off; off += (size_t)B * D;
    float* u     = ws + off; off += (size_t)B * 4 * D;
    float* qku   = ws + off; off += (size_t)B * 4 * D;
    float* squ   = ws + off; off += (size_t)B * 4 * UH;
    float* sku   = ws + off; off += (size_t)B * 4 * UH;
    float* u2    = ws + off; off += (size_t)B * 4 * D;
    float* h1    = ws + off; off += (size_t)4 * B * FH;
    float* h2    = ws + off; off += (size_t)4 * B * FH;
    (void)ws_size; (void)n_in; (void)in_sizes; (void)out_size;

    // ---- P_GAT, 2 layers ----
    const float* cur = feature;
    float* nxt = feat0;
    for (int layer = 0; layer < 2; ++layer) {
        gemm16_wmma<<<dim3((B * N) / 16, D / 32), 32, 0, stream>>>(
            cur, pgat_W, nullptr, qk, B * N, D, D);
        pgat_sqsk<<<(B * N * PH + 255) / 256, 256, 0, stream>>>(qk, pgat_attn, sq, sk);
        pgat_attn_step<<<dim3(N / 16, PH, B), 32, 0, stream>>>(
            cur, qk, sq, sk, p_mask, nxt);
        cur = nxt;
        nxt = feat1;
    }
    const float* new_p = feat1;

    // ---- comb = concat(new_p, c_node) ----
    build_comb<<<((size_t)B * NP1 * D + 255) / 256, 256, 0, stream>>>(new_p, c_node, comb);

    // ---- 4x UP_GAT (sequential, reuse buffers) ----
    for (int i = 0; i < 4; ++i) {
        gemm16_wmma<<<dim3(B / 16, D / 32), 32, 0, stream>>>(
            c_node, up_qw[i], up_qb[i], qbuf, B, D, D);
        gemm16_wmma<<<dim3((B * NP1) / 16, D / 32), 32, 0, stream>>>(
            comb, up_kw[i], up_kb[i], qk, B * NP1, D, D);
        up_sim<<<(B * NP1 + 255) / 256, 256, 0, stream>>>(qbuf, qk, p_mask, sim);
        up_softmax<<<B, 256, 0, stream>>>(sim);
        up_head<<<(B * D + 255) / 256, 256, 0, stream>>>(sim, comb, headb);
        ow_eff_k<<<(D * D + 255) / 256, 256, 0, stream>>>(up_ow[i], oweff);
        // scatter rows: row b lands at u[(b*4 + i)*D]
        gemm16_wmma<<<dim3(B / 16, D / 32), 32, 0, stream>>>(
            headb, oweff, up_ob[i], u + (size_t)i * D, B, D, 4 * D);
    }

    // ---- U_GAT (1 layer, 4 nodes, no adjacency) ----
    gemm16_wmma<<<dim3((B * 4) / 16, D / 32), 32, 0, stream>>>(
        u, ugat_W, nullptr, qku, B * 4, D, D);
    ugat_sqsk<<<1, 256, 0, stream>>>(qku, ugat_attn, squ, sku);
    u_attn<<<(B * 4 * UH * DHU + 255) / 256, 256, 0, stream>>>(u, qku, squ, sku, u2);

    // ---- MLP + classification heads ----
    mlp1_k<<<(4 * B * FH + 255) / 256, 256, 0, stream>>>(u2, mlp_w1, mlp_b1, h1);
    mlp2_k<<<(4 * B * FH + 255) / 256, 256, 0, stream>>>(h1, mlp_w2, mlp_b2, h2);
    heads_k<<<1, 128, 0, stream>>>(h2, fcf_w[0], fcf_b[0], fcf_w[1], fcf_b[1],
                                   fcf_w[2], fcf_b[2], fcf_w[3], fcf_b[3], out);
}